// RGCNLayer_45784351375471
// MI455X (gfx1250) — compile-verified
//
#include <hip/hip_runtime.h>

#define NN 10000
#define NE 160000
#define DIN 128
#define DOUT 128
#define NB 8

typedef __attribute__((ext_vector_type(2))) float v2f;
typedef __attribute__((ext_vector_type(8))) float v8f;

__device__ __forceinline__ v8f wmma_f32_k4(v2f a, v2f b, v8f c) {
  // D = A(16x4 f32) * B(4x16 f32) + C, exact fp32 -> matches reference math
  return __builtin_amdgcn_wmma_f32_16x16x4_f32(false, a, false, b, (short)0, c,
                                               false, false);
}

// ---------------------------------------------------------------------------
// Kernel 1: A16[e,0:8] = edge_embs @ W_rb ; A16[e,8:16] = edge_embs @ W_rb_inv
// 4 waves/block, 1 wave per 16-edge tile.
//  - B panel (128x16 concat) staged once per block in LDS (branchless).
//  - Each wave stages its 16x128 edge tile via dense 512B/instr b128 loads,
//    padded to 132 floats/row -> conflict-free ds_load_b64 A fragments.
// ---------------------------------------------------------------------------
__global__ __launch_bounds__(128) void coef_gemm_kernel(
    const float* __restrict__ E, const float* __restrict__ Wrb,
    const float* __restrict__ Wrbi, float* __restrict__ A16) {
  __shared__ __align__(16) float sB[DIN * 16];      // [k][col], 8 KB
  __shared__ __align__(16) float sE[4][16 * 132];   // per-wave edge tile

  const int tid = threadIdx.x;
  // Cooperative branchless B-panel stage: 2048 floats / 128 threads.
#pragma unroll
  for (int i = tid; i < DIN * 16; i += 128) {
    const int k = i >> 4;
    const int c = i & 15;
    sB[i] = (c < 8) ? Wrb[k * 8 + c] : Wrbi[k * 8 + (c - 8)];
  }

  const int wv   = tid >> 5;
  const int lane = tid & 31;
  const int tile = blockIdx.x * 4 + wv;  // 2500 blocks * 4 = 10000 tiles
  float* myE = sE[wv];
  const float* gsrc = E + (size_t)tile * 16 * DIN;
  // Dense stage: instruction t moves one full 512B row (32 lanes x b128).
#pragma unroll
  for (int t = 0; t < 16; ++t) {
    const float4 v = *(const float4*)(gsrc + t * DIN + lane * 4);
    *(float4*)(myE + t * 132 + lane * 4) = v;
  }
  __syncthreads();

  const int row   = lane & 15;  // A row (edge) and B column
  const int khalf = lane >> 4;  // 0: K={0,1}, 1: K={2,3}
  const float* arow = myE + row * 132;
  v8f c = {};
#pragma unroll
  for (int k0 = 0; k0 < DIN; k0 += 4) {
    const int ka = k0 + khalf * 2;
    v2f a;
    a.x = arow[ka];
    a.y = arow[ka + 1];
    v2f b;
    b.x = sB[ka * 16 + row];
    b.y = sB[(ka + 1) * 16 + row];
    c = wmma_f32_k4(a, b, c);
  }
#pragma unroll
  for (int r = 0; r < 8; ++r) {
    const int m = r + khalf * 8;  // C/D layout: VGPR r -> M = r (+8 hi half)
    A16[(size_t)(tile * 16 + m) * 16 + row] = c[r];
  }
}

// ---------------------------------------------------------------------------
// Kernel 2: Z[n,b,:] = node_embs @ W_basis[b]  (bslot 0..7)
//           Hself[n,:] = node_embs @ W_self    (bslot 8)
// grid = (625 node tiles, 9 slots); 8 waves/block cover the 128 output cols.
// X tile is reused by all 8 waves -> staged once in LDS. W read from global
// (each element used once per block; L2-resident across the 625 tiles).
// ---------------------------------------------------------------------------
__global__ __launch_bounds__(256) void basis_gemm_kernel(
    const float* __restrict__ X, const float* __restrict__ Wb,
    const float* __restrict__ Wself, float* __restrict__ Z,
    float* __restrict__ Hself) {
  __shared__ __align__(16) float sX[16 * 132];  // 16 nodes x 128 (+4 pad)

  const int tid   = threadIdx.x;
  const int ntile = blockIdx.x;  // 625 tiles * 16 = 10000 nodes exactly
  const int bslot = blockIdx.y;  // 0..7 basis, 8 = W_self

  {  // cooperative dense stage: thread -> 8 consecutive floats (2 x b128)
    const int r = tid >> 4;
    const int cc = (tid & 15) * 8;
    const float* g = X + (size_t)(ntile * 16 + r) * DIN + cc;
    *(float4*)(sX + r * 132 + cc)     = *(const float4*)(g);
    *(float4*)(sX + r * 132 + cc + 4) = *(const float4*)(g + 4);
  }
  __syncthreads();

  const int lane  = tid & 31;
  const int wv    = tid >> 5;  // 0..7 -> which 16-col tile
  const int row   = lane & 15;
  const int khalf = lane >> 4;
  const float* W =
      (bslot < NB) ? (Wb + (size_t)bslot * DIN * DOUT) : Wself;  // [k][o]
  const int col = wv * 16 + row;
  const float* arow = sX + row * 132;

  v8f c = {};
#pragma unroll
  for (int k0 = 0; k0 < DIN; k0 += 4) {
    const int ka = k0 + khalf * 2;
    v2f a;
    a.x = arow[ka];
    a.y = arow[ka + 1];
    v2f b;
    b.x = W[(size_t)ka * DOUT + col];
    b.y = W[(size_t)(ka + 1) * DOUT + col];
    c = wmma_f32_k4(a, b, c);
  }
#pragma unroll
  for (int r = 0; r < 8; ++r) {
    const int node = ntile * 16 + r + khalf * 8;
    if (bslot < NB)
      Z[(size_t)node * (NB * DOUT) + bslot * DOUT + col] = c[r];
    else
      Hself[(size_t)node * DOUT + col] = c[r];
  }
}

// ---------------------------------------------------------------------------
// Kernel 3: per-edge message = 8-coeff combination of L2-resident Z rows,
// atomic mean-scatter (fwd: Z[src] -> dst, rev: Z[dst] -> src).
// One wave per edge; edge id forced wave-uniform so coefficient/index
// fetches become scalar (SMEM) loads.
// ---------------------------------------------------------------------------
__global__ __launch_bounds__(256) void edge_scatter_kernel(
    const float* __restrict__ A16, const float* __restrict__ Z,
    const int* __restrict__ src, const int* __restrict__ dst,
    float* __restrict__ sumF, float* __restrict__ sumR,
    float* __restrict__ cntF, float* __restrict__ cntR) {
  const int e =
      __builtin_amdgcn_readfirstlane((blockIdx.x * 256 + threadIdx.x) >> 5);
  const int lane = threadIdx.x & 31;
  if (e >= NE) return;

  const int s = src[e];
  const int d = dst[e];
  const float* a = A16 + (size_t)e * 16;
  float af[NB], ar[NB];
#pragma unroll
  for (int b = 0; b < NB; ++b) {
    af[b] = a[b];
    ar[b] = a[NB + b];
  }
  const float* Zs = Z + (size_t)s * (NB * DOUT);
  const float* Zd = Z + (size_t)d * (NB * DOUT);
  const int o = lane * 4;
  float mf0 = 0.f, mf1 = 0.f, mf2 = 0.f, mf3 = 0.f;
  float mr0 = 0.f, mr1 = 0.f, mr2 = 0.f, mr3 = 0.f;
#pragma unroll
  for (int b = 0; b < NB; ++b) {
    const float4 zs = *(const float4*)(Zs + b * DOUT + o);
    const float4 zd = *(const float4*)(Zd + b * DOUT + o);
    mf0 = fmaf(af[b], zs.x, mf0);
    mf1 = fmaf(af[b], zs.y, mf1);
    mf2 = fmaf(af[b], zs.z, mf2);
    mf3 = fmaf(af[b], zs.w, mf3);
    mr0 = fmaf(ar[b], zd.x, mr0);
    mr1 = fmaf(ar[b], zd.y, mr1);
    mr2 = fmaf(ar[b], zd.z, mr2);
    mr3 = fmaf(ar[b], zd.w, mr3);
  }
  float* pf = sumF + (size_t)d * DOUT + o;
  float* pr = sumR + (size_t)s * DOUT + o;
  unsafeAtomicAdd(pf + 0, mf0);
  unsafeAtomicAdd(pf + 1, mf1);
  unsafeAtomicAdd(pf + 2, mf2);
  unsafeAtomicAdd(pf + 3, mf3);
  unsafeAtomicAdd(pr + 0, mr0);
  unsafeAtomicAdd(pr + 1, mr1);
  unsafeAtomicAdd(pr + 2, mr2);
  unsafeAtomicAdd(pr + 3, mr3);
  if (lane == 0) {
    unsafeAtomicAdd(cntF + d, 1.0f);
    unsafeAtomicAdd(cntR + s, 1.0f);
  }
}

// ---------------------------------------------------------------------------
// Kernel 4: out = relu(sumF/max(cF,1) + sumR/max(cR,1) + Hself + b_self)
// ---------------------------------------------------------------------------
__global__ __launch_bounds__(256) void finalize_kernel(
    const float* __restrict__ sumF, const float* __restrict__ sumR,
    const float* __restrict__ cntF, const float* __restrict__ cntR,
    const float* __restrict__ Hself, const float* __restrict__ bself,
    float* __restrict__ out) {
  const int i = blockIdx.x * 256 + threadIdx.x;
  if (i >= NN * DOUT) return;
  const int n = i >> 7;
  const int o = i & (DOUT - 1);
  const float cf = fmaxf(cntF[n], 1.0f);
  const float cr = fmaxf(cntR[n], 1.0f);
  const float v = sumF[i] / cf + sumR[i] / cr + Hself[i] + bself[o];
  out[i] = fmaxf(v, 0.0f);
}

extern "C" void kernel_launch(void* const* d_in, const int* in_sizes, int n_in,
                              void* d_out, int out_size, void* d_ws,
                              size_t ws_size, hipStream_t stream) {
  (void)in_sizes; (void)n_in; (void)out_size; (void)ws_size;
  const float* node  = (const float*)d_in[0];
  const float* edge  = (const float*)d_in[1];
  const int*   src   = (const int*)d_in[2];
  const int*   dst   = (const int*)d_in[3];
  const float* Wrb   = (const float*)d_in[4];
  const float* Wrbi  = (const float*)d_in[5];
  const float* Wb    = (const float*)d_in[6];
  const float* Wself = (const float*)d_in[7];
  const float* bself = (const float*)d_in[8];
  float* out = (float*)d_out;

  // Workspace layout (floats): total 16,660,000 (~66.6 MB)
  float* ws    = (float*)d_ws;
  float* Z     = ws;                               // 10,240,000 (NN*NB*DOUT)
  float* Hself = Z + (size_t)NN * NB * DOUT;       // 1,280,000
  float* A16   = Hself + (size_t)NN * DOUT;        // 2,560,000 (NE*16)
  float* sumF  = A16 + (size_t)NE * 16;            // 1,280,000
  float* sumR  = sumF + (size_t)NN * DOUT;         // 1,280,000
  float* cntF  = sumR + (size_t)NN * DOUT;         // 10,000
  float* cntR  = cntF + NN;                        // 10,000

  // Zero accumulators + counters every call (deterministic re-init).
  hipMemsetAsync(sumF, 0,
                 ((size_t)2 * NN * DOUT + 2 * NN) * sizeof(float), stream);

  // 10000 edge tiles -> 2500 blocks of 4 waves
  coef_gemm_kernel<<<2500, 128, 0, stream>>>(edge, Wrb, Wrbi, A16);
  // 625 node tiles x 9 slots (8 bases + self)
  basis_gemm_kernel<<<dim3(625, 9), 256, 0, stream>>>(node, Wb, Wself, Z,
                                                      Hself);
  // 160000 waves -> 20000 blocks
  edge_scatter_kernel<<<20000, 256, 0, stream>>>(A16, Z, src, dst, sumF, sumR,
                                                 cntF, cntR);
  finalize_kernel<<<(NN * DOUT + 255) / 256, 256, 0, stream>>>(
      sumF, sumR, cntF, cntR, Hself, bself, out);
}